// AdversarialLoss_PDD_90211493085246
// MI455X (gfx1250) — compile-verified
//
#include <hip/hip_runtime.h>
#include <hip/hip_bf16.h>
#include <math.h>

// Problem constants (from reference)
#define BN   192          // batch per domain
#define DIM  2048         // feature dim
#define NC   1000         // classes
#define MROWS 384         // 2*BN
#define TEMPSQ 4.0f
#define THRESH 0.001f

typedef float v2f __attribute__((ext_vector_type(2)));
typedef float v8f __attribute__((ext_vector_type(8)));

// ---------------------------------------------------------------------------
// Kernel 1: y[384 x 1000] = f[384 x 2048] @ W[2048 x 1000] + b
// Workgroup = 2 waves. Each wave owns half of K (1024) for a 16x64 output
// tile: 4 N-subtiles share one A fragment (A reuse 4x), 4 accumulators,
// V_WMMA_F32_16X16X4_F32. Partials reduced across the 2 waves through LDS.
// ---------------------------------------------------------------------------
__global__ __launch_bounds__(64)
void gemm_wmma_f32(const float* __restrict__ f, const float* __restrict__ W,
                   const float* __restrict__ bias, float* __restrict__ y)
{
    const int tid  = threadIdx.x;
    const int wave = tid >> 5;          // 0..1 : K-split
    const int lane = tid & 31;
    const int half = lane >> 4;         // 0/1 : K pair within step / row group
    const int l    = lane & 15;
    const int m0   = blockIdx.x * 16;   // 24 M tiles
    const int n0   = blockIdx.y * 64;   // 16 N tiles (last partial)
    const int KH   = DIM / 2;           // K per wave

    // A: lane l holds row m0+l; VGPR0/1 = K = 2*half, 2*half+1 (+ wave's K base)
    const float* arow = f + (size_t)(m0 + l) * DIM + 2 * half + wave * KH;

    // B: 4 N-subtiles; lane l holds column n0 + 16*a + l
    bool ok[4];
    const float* bcol[4];
    #pragma unroll
    for (int a = 0; a < 4; ++a) {
        const int col = n0 + a * 16 + l;
        ok[a] = (col < NC);
        const int colc = ok[a] ? col : (NC - 1);          // clamped address
        bcol[a] = W + colc + (size_t)(2 * half + wave * KH) * NC;
    }

    v8f acc[4] = {};
    #pragma unroll 2
    for (int k = 0; k < KH; k += 4) {
        v2f afr;
        afr.x = arow[k];
        afr.y = arow[k + 1];
        #pragma unroll
        for (int a = 0; a < 4; ++a) {
            float b0 = bcol[a][(size_t)k * NC];           // loads unconditional,
            float b1 = bcol[a][(size_t)(k + 1) * NC];     // value zero-selected:
            v2f bfr;                                      // EXEC stays all-1s
            bfr.x = ok[a] ? b0 : 0.0f;
            bfr.y = ok[a] ? b1 : 0.0f;
            acc[a] = __builtin_amdgcn_wmma_f32_16x16x4_f32(
                         false, afr, false, bfr, (short)0, acc[a], false, false);
        }
    }

    // Reduce the two K-halves through LDS, then store with bias.
    __shared__ float red[2][4][8][32];
    #pragma unroll
    for (int a = 0; a < 4; ++a)
        #pragma unroll
        for (int r = 0; r < 8; ++r)
            red[wave][a][r][lane] = acc[a][r];
    __syncthreads();

    // 16 rows x 64 cols; thread t handles column t of each row (coalesced).
    #pragma unroll
    for (int row = 0; row < 16; ++row) {
        const int col  = tid;                 // 0..63
        const int aa   = col >> 4, ll = col & 15;
        const int hh   = row >> 3, rr = row & 7;
        const float s  = red[0][aa][rr][hh * 16 + ll] + red[1][aa][rr][hh * 16 + ll];
        const int gcol = n0 + col;
        if (gcol < NC)
            y[(size_t)(m0 + row) * NC + gcol] = s + bias[gcol];
    }
}

// ---------------------------------------------------------------------------
// Kernel 2: per-row stats. lse2 = logsumexp(y/2) for all rows.
// For t rows (>=192): labels_t = argmax(y) (first max), conf = 1/sum(exp(y-max)).
// ---------------------------------------------------------------------------
__global__ __launch_bounds__(256)
void row_stats(const float* __restrict__ y, float* __restrict__ lse2,
               float* __restrict__ conf, int* __restrict__ labels_t)
{
    const int row = blockIdx.x;
    const int tid = threadIdx.x;
    const float* yr = y + (size_t)row * NC;

    __shared__ float smax[256];
    __shared__ int   sidx[256];

    float bm = -INFINITY; int bi = NC;
    for (int c = tid; c < NC; c += 256) {
        float v = yr[c];
        if (v > bm || (v == bm && c < bi)) { bm = v; bi = c; }
    }
    smax[tid] = bm; sidx[tid] = bi;
    __syncthreads();
    for (int s = 128; s > 0; s >>= 1) {
        if (tid < s) {
            float v = smax[tid + s]; int i2 = sidx[tid + s];
            if (v > smax[tid] || (v == smax[tid] && i2 < sidx[tid])) {
                smax[tid] = v; sidx[tid] = i2;
            }
        }
        __syncthreads();
    }
    const float m  = smax[0];
    const int   am = sidx[0];

    __shared__ float sh1[256], sh2[256];
    float s1 = 0.0f, s2 = 0.0f;
    for (int c = tid; c < NC; c += 256) {
        float v = yr[c] - m;
        s1 += __expf(v);
        s2 += __expf(0.5f * v);
    }
    sh1[tid] = s1; sh2[tid] = s2;
    __syncthreads();
    for (int s = 128; s > 0; s >>= 1) {
        if (tid < s) { sh1[tid] += sh1[tid + s]; sh2[tid] += sh2[tid + s]; }
        __syncthreads();
    }
    if (tid == 0) {
        lse2[row] = 0.5f * m + __logf(sh2[0]);
        if (row >= BN) {
            conf[row - BN]     = 1.0f / sh1[0];  // softmax at its own argmax
            labels_t[row - BN] = am;
        }
    }
}

// ---------------------------------------------------------------------------
// Block-sum helper (fixed tree -> deterministic ordering)
// ---------------------------------------------------------------------------
__device__ __forceinline__ float block_sum_256(float v, float* sh)
{
    const int tid = threadIdx.x;
    sh[tid] = v;
    __syncthreads();
    for (int s = 128; s > 0; s >>= 1) {
        if (tid < s) sh[tid] += sh[tid + s];
        __syncthreads();
    }
    return sh[0];
}

// ---------------------------------------------------------------------------
// Kernel 3a: JS(s_i, t_j) for masked pairs only. Unmasked -> write 0 and exit.
// ---------------------------------------------------------------------------
__global__ __launch_bounds__(256)
void pair_js_st(const float* __restrict__ y, const float* __restrict__ lse2,
                const float* __restrict__ conf, const int* __restrict__ labels_t,
                const int* __restrict__ labels_s, float* __restrict__ js_st)
{
    const int i = blockIdx.x, j = blockIdx.y;
    const int tid = threadIdx.x;
    const bool mask = (conf[j] >= THRESH) && (labels_s[i] == labels_t[j]);
    if (!mask) { if (tid == 0) js_st[i * BN + j] = 0.0f; return; }

    const float* ys = y + (size_t)i * NC;
    const float* yt = y + (size_t)(BN + j) * NC;
    const float  li = lse2[i];
    const float  lj = lse2[BN + j];

    float acc = 0.0f;
    for (int c = tid; c < NC; c += 256) {
        float lp = 0.5f * ys[c] - li;
        float lq = 0.5f * yt[c] - lj;
        float p = __expf(lp), q = __expf(lq);
        float lm = __logf(0.5f * (p + q));
        acc += p * (lp - lm) + q * (lq - lm);
    }
    __shared__ float sh[256];
    float tot = block_sum_256(acc, sh);
    if (tid == 0) js_st[i * BN + j] = 0.5f * tot;
}

// ---------------------------------------------------------------------------
// Kernel 3b: JS(s_i, s_j) for i<j masked pairs only.
// ---------------------------------------------------------------------------
__global__ __launch_bounds__(256)
void pair_js_ss(const float* __restrict__ y, const float* __restrict__ lse2,
                const int* __restrict__ labels_s, float* __restrict__ js_ss)
{
    const int i = blockIdx.x, j = blockIdx.y;
    const int tid = threadIdx.x;
    const bool mask = (i < j) && (labels_s[i] == labels_s[j]);
    if (!mask) { if (tid == 0) js_ss[i * BN + j] = 0.0f; return; }

    const float* yi = y + (size_t)i * NC;
    const float* yj = y + (size_t)j * NC;
    const float  li = lse2[i];
    const float  lj = lse2[j];

    float acc = 0.0f;
    for (int c = tid; c < NC; c += 256) {
        float lp = 0.5f * yi[c] - li;
        float lq = 0.5f * yj[c] - lj;
        float p = __expf(lp), q = __expf(lq);
        float lm = __logf(0.5f * (p + q));
        acc += p * (lp - lm) + q * (lq - lm);
    }
    __shared__ float sh[256];
    float tot = block_sum_256(acc, sh);
    if (tid == 0) js_ss[i * BN + j] = 0.5f * tot;
}

// ---------------------------------------------------------------------------
// Kernel 4: recompute masks cheaply, reduce sums/counts, emit scalar loss.
// ---------------------------------------------------------------------------
__global__ __launch_bounds__(256)
void finalize(const float* __restrict__ js_st, const float* __restrict__ js_ss,
              const float* __restrict__ conf, const int* __restrict__ labels_t,
              const int* __restrict__ labels_s, float* __restrict__ out)
{
    const int tid = threadIdx.x;
    float sst = 0.0f, sss = 0.0f;
    int   cst = 0,    css = 0;
    for (int idx = tid; idx < BN * BN; idx += 256) {
        const int i = idx / BN, j = idx % BN;
        if ((conf[j] >= THRESH) && (labels_s[i] == labels_t[j])) {
            sst += js_st[idx]; cst++;
        }
        if ((i < j) && (labels_s[i] == labels_s[j])) {
            sss += js_ss[idx]; css++;
        }
    }
    __shared__ float shf[256];
    __shared__ int   shi[256];
    float Sst = block_sum_256(sst, shf);
    __syncthreads();
    float Sss = block_sum_256(sss, shf);
    shi[tid] = cst; __syncthreads();
    for (int s = 128; s > 0; s >>= 1) { if (tid < s) shi[tid] += shi[tid + s]; __syncthreads(); }
    const int Cst = shi[0]; __syncthreads();
    shi[tid] = css; __syncthreads();
    for (int s = 128; s > 0; s >>= 1) { if (tid < s) shi[tid] += shi[tid + s]; __syncthreads(); }
    const int Css = shi[0];

    if (tid == 0) {
        float lst = (Cst > 0) ? (Sst / (float)Cst) : 0.0f;
        float lss = (Css > 0) ? (Sss / (float)Css) : 0.0f;
        out[0] = TEMPSQ * (lst + lss);
    }
}

// ---------------------------------------------------------------------------
// Launch
// ---------------------------------------------------------------------------
extern "C" void kernel_launch(void* const* d_in, const int* in_sizes, int n_in,
                              void* d_out, int out_size, void* d_ws, size_t ws_size,
                              hipStream_t stream)
{
    const float* f        = (const float*)d_in[0];   // (384, 2048)
    const float* W        = (const float*)d_in[1];   // (2048, 1000)
    const float* b        = (const float*)d_in[2];   // (1000,)
    const int*   labels_s = (const int*)d_in[3];     // (192,)
    float* out = (float*)d_out;

    // Workspace layout (floats)
    float* ws    = (float*)d_ws;
    float* y     = ws;                                  // 384*1000
    float* lse2  = y     + (size_t)MROWS * NC;          // 384
    float* conf  = lse2  + MROWS;                       // 192
    int*   lab_t = (int*)(conf + BN);                   // 192
    float* js_st = (float*)(lab_t + BN);                // 192*192
    float* js_ss = js_st + (size_t)BN * BN;             // 192*192

    dim3 gGemm(MROWS / 16, (NC + 63) / 64);             // 24 x 16 tiles of 16x64
    gemm_wmma_f32<<<gGemm, 64, 0, stream>>>(f, W, b, y);

    row_stats<<<MROWS, 256, 0, stream>>>(y, lse2, conf, lab_t);

    dim3 gPair(BN, BN);
    pair_js_st<<<gPair, 256, 0, stream>>>(y, lse2, conf, lab_t, labels_s, js_st);
    pair_js_ss<<<gPair, 256, 0, stream>>>(y, lse2, labels_s, js_ss);

    finalize<<<1, 256, 0, stream>>>(js_st, js_ss, conf, lab_t, labels_s, out);
}